// DiscreteDistributionOutput_7069516169439
// MI455X (gfx1250) — compile-verified
//
#include <hip/hip_runtime.h>

typedef __attribute__((ext_vector_type(2))) float v2f;
typedef __attribute__((ext_vector_type(4))) float v4f;
typedef __attribute__((ext_vector_type(8))) float v8f;

#define B_       32
#define CIN      64
#define PC_      3
#define KC       64
#define HW_      16384
#define NTILE    5
#define NPAIR    15
#define NCHUNK   32
#define CHUNK_HW (HW_/NCHUNK)      /* 512 */
#define ITERS    (CHUNK_HW/16)     /* 32  */
#define TILE_ELEMS 256
#define SPART_STRIDE (NPAIR*TILE_ELEMS)  /* 3840 floats per (b,chunk) */

/* d_out layout (flat, return order) */
#define OUT_DM   1572864   /* distance_matrix (32x64) */
#define OUT_IDX  1574912   /* idx_k (32)              */
#define OUT_LOSS 1574944   /* loss (32)               */
#define OUT_DIST 1574976   /* distances (32)          */

__device__ __forceinline__ int pair_idx(int i, int j) { /* i<=j<=3 */
  return i*(7-i)/2 + j;
}

__device__ __forceinline__ void load8(const float* __restrict__ p, float* a) {
  v4f lo = *(const v4f*)p;
  v4f hi = *(const v4f*)(p + 4);
  a[0]=lo.x; a[1]=lo.y; a[2]=lo.z; a[3]=lo.w;
  a[4]=hi.x; a[5]=hi.y; a[6]=hi.z; a[7]=hi.w;
}

/* Pass 1: augmented Gram via fp32 WMMA.
 * Rows 0..63 = feat channels (tiles 0..3); tile 4 rows 0..2 = predict_last-target,
 * rows 3..15 = 0. Accumulate 15 symmetric tile pairs over a 512-wide hw chunk. */
__global__ void __launch_bounds__(32)
gram_kernel(const float* __restrict__ feat, const float* __restrict__ target,
            const float* __restrict__ plast, float* __restrict__ spart)
{
  const int b     = blockIdx.x;
  const int chunk = blockIdx.y;
  const int lane  = threadIdx.x;
  const int g = lane >> 4;       /* lane group: selects hw sub-block + D row half */
  const int r = lane & 15;       /* row within 16-row tile / column of D          */

  const float* fb = feat   + (size_t)b * CIN * HW_;
  const float* tb = target + (size_t)b * PC_ * HW_;
  const float* pb = plast  + (size_t)b * PC_ * HW_;

  v8f zero = {};
  v8f acc[NPAIR];
  #pragma unroll
  for (int p = 0; p < NPAIR; ++p) acc[p] = zero;

  const int   rp = (r < PC_) ? r : (PC_ - 1);
  const float rs = (r < PC_) ? 1.0f : 0.0f;
  const int hwbase = chunk * CHUNK_HW + g * 8;

  static constexpr int PI[NPAIR] = {0,0,0,0,1,1,1,2,2,3, 0,1,2,3, 4};
  static constexpr int PJ[NPAIR] = {0,1,2,3,1,2,3,2,3,3, 4,4,4,4, 4};

  for (int it = 0; it < ITERS; ++it) {
    const int hw = hwbase + it * 16;
    float a[NTILE][8];
    #pragma unroll
    for (int t = 0; t < 4; ++t)
      load8(fb + (size_t)(t*16 + r) * HW_ + hw, a[t]);
    {
      float pl[8], tg[8];
      load8(pb + (size_t)rp * HW_ + hw, pl);
      load8(tb + (size_t)rp * HW_ + hw, tg);
      #pragma unroll
      for (int e = 0; e < 8; ++e) a[4][e] = (pl[e] - tg[e]) * rs;
    }
    /* 4 WMMA K-steps; lane-group g supplies hw offsets {g*8+2s, g*8+2s+1} in
     * K-slots {2g, 2g+1} — identical mapping for A and B operands. */
    #pragma unroll
    for (int s = 0; s < 4; ++s) {
      v2f op[NTILE];
      #pragma unroll
      for (int t = 0; t < NTILE; ++t) { op[t].x = a[t][2*s]; op[t].y = a[t][2*s+1]; }
      #pragma unroll
      for (int p = 0; p < NPAIR; ++p)
        acc[p] = __builtin_amdgcn_wmma_f32_16x16x4_f32(
            false, op[PI[p]], false, op[PJ[p]], (short)0, acc[p], false, false);
    }
  }

  /* D layout: lane group g holds rows M = g*8 + v, column N = r. */
  float* out = spart + ((size_t)b * NCHUNK + chunk) * SPART_STRIDE;
  #pragma unroll
  for (int p = 0; p < NPAIR; ++p) {
    #pragma unroll
    for (int v = 0; v < 8; ++v)
      out[p*TILE_ELEMS + (g*8 + v)*16 + r] = acc[p][v];
  }
}

/* Pass 2: deterministic chunk reduction, distance matrix, argmin. */
__global__ void __launch_bounds__(64)
distance_kernel(const float* __restrict__ spart, const float* __restrict__ weight,
                float* __restrict__ dout, int* __restrict__ idxbuf)
{
  __shared__ float S[SPART_STRIDE];      /* reduced Gram tiles, 15 KB */
  __shared__ float W[KC*PC_*CIN];        /* full weight, 48 KB        */
  __shared__ float dsh[KC];

  const int b   = blockIdx.x;
  const int tid = threadIdx.x;

  for (int e = tid; e < SPART_STRIDE; e += 64) {
    const float* p = spart + (size_t)b * NCHUNK * SPART_STRIDE + e;
    float s = 0.f;
    for (int ch = 0; ch < NCHUNK; ++ch) s += p[(size_t)ch * SPART_STRIDE];
    S[e] = s;
  }
  for (int e = tid; e < KC*PC_*CIN; e += 64) W[e] = weight[e];
  __syncthreads();

  const float rsum = S[14*TILE_ELEMS + 0] + S[14*TILE_ELEMS + 17] + S[14*TILE_ELEMS + 34];

  const int k = tid;                 /* one candidate per thread */
  float d = 0.f;
  for (int p = 0; p < PC_; ++p) {
    const float* w = &W[(k*PC_ + p) * CIN];
    float t1 = 0.f, t2 = 0.f;
    for (int c = 0; c < CIN; ++c) {
      const int ti = c >> 4, ri = c & 15;
      float h = 0.f;
      for (int c2 = 0; c2 < CIN; ++c2) {
        const int tj = c2 >> 4, rj = c2 & 15;
        const float gv = (ti <= tj)
            ? S[pair_idx(ti, tj)*TILE_ELEMS + ri*16 + rj]
            : S[pair_idx(tj, ti)*TILE_ELEMS + rj*16 + ri];
        h += gv * w[c2];
      }
      t1 += w[c] * h;
      t2 += w[c] * S[(10 + ti)*TILE_ELEMS + ri*16 + p];
    }
    d += t1 + 2.f * t2;
  }
  d = (d + rsum) * (1.0f / (PC_ * HW_));
  dsh[k] = d;
  dout[OUT_DM + b*KC + k] = d;
  __syncthreads();

  if (tid == 0) {
    float best = dsh[0]; int bi = 0;
    for (int i = 1; i < KC; ++i) if (dsh[i] < best) { best = dsh[i]; bi = i; }
    dout[OUT_IDX  + b] = (float)bi;
    dout[OUT_LOSS + b] = best;   /* loss == distance at argmin (identical math) */
    dout[OUT_DIST + b] = best;
    idxbuf[b] = bi;
  }
}

/* Pass 3: compute only the selected 3 output channels (feat is L2-hot). */
__global__ void __launch_bounds__(256)
predict_kernel(const float* __restrict__ feat, const float* __restrict__ plast,
               const float* __restrict__ weight, const int* __restrict__ idxbuf,
               float* __restrict__ dout)
{
  __shared__ float wl[PC_*CIN];
  const int blk   = blockIdx.x;
  const int b     = blk >> 6;
  const int chunk = blk & 63;
  const int tid   = threadIdx.x;

  if (tid < PC_*CIN) {
    const int k = idxbuf[b];
    wl[tid] = weight[(size_t)(k * PC_) * CIN + tid];
  }
  __syncthreads();

  const int hw = chunk*256 + tid;
  const float* fb = feat + (size_t)b * CIN * HW_ + hw;
  float a0 = 0.f, a1 = 0.f, a2 = 0.f;
  #pragma unroll 8
  for (int c = 0; c < CIN; ++c) {
    const float f = fb[(size_t)c * HW_];
    a0 += f * wl[0*CIN + c];
    a1 += f * wl[1*CIN + c];
    a2 += f * wl[2*CIN + c];
  }
  const float* pb = plast + (size_t)b * PC_ * HW_ + hw;
  float* ob = dout + (size_t)b * PC_ * HW_ + hw;
  ob[0]     = a0 + pb[0];
  ob[HW_]   = a1 + pb[HW_];
  ob[2*HW_] = a2 + pb[2*HW_];
}

extern "C" void kernel_launch(void* const* d_in, const int* in_sizes, int n_in,
                              void* d_out, int out_size, void* d_ws, size_t ws_size,
                              hipStream_t stream)
{
  (void)in_sizes; (void)n_in; (void)out_size; (void)ws_size;
  const float* feat   = (const float*)d_in[0];  /* (32,64,128,128)  */
  const float* target = (const float*)d_in[1];  /* (32,3,128,128)   */
  const float* plast  = (const float*)d_in[2];  /* (32,3,128,128)   */
  const float* weight = (const float*)d_in[3];  /* (192,64)         */
  float* out = (float*)d_out;

  float* spart = (float*)d_ws;  /* 32*32*3840 floats = 15.7 MB */
  int* idxbuf  = (int*)((char*)d_ws + (size_t)B_ * NCHUNK * SPART_STRIDE * sizeof(float));

  gram_kernel<<<dim3(B_, NCHUNK), 32, 0, stream>>>(feat, target, plast, spart);
  distance_kernel<<<B_, 64, 0, stream>>>(spart, weight, out, idxbuf);
  predict_kernel<<<B_ * 64, 256, 0, stream>>>(feat, plast, weight, idxbuf, out);
}